// SimpleRGCNLayer_72756745994393
// MI455X (gfx1250) — compile-verified
//
#include <hip/hip_runtime.h>

#define HD 128

typedef __attribute__((ext_vector_type(2))) float v2f;
typedef __attribute__((ext_vector_type(4))) float v4f;
typedef __attribute__((ext_vector_type(8))) float v8f;

// A-operand staging: per wave, per 16-row tile, entries [lane'][kk] (lane' = half*16+m),
// v2f with row stride 33 -> bank = (2*lane + 2*kk) % 64, conflict-free for loads & stores.
#define ASTRIDE 33
#define ATILE (32 * ASTRIDE)  // v2f elements per 16-row tile

// ---------------- utility kernels ----------------

__global__ void k_zero_f32(float* __restrict__ p, int n) {
  int i = blockIdx.x * blockDim.x + threadIdx.x;
  if (i < n) p[i] = 0.0f;
}

__global__ void k_zero_i32(int* __restrict__ p, int n) {
  int i = blockIdx.x * blockDim.x + threadIdx.x;
  if (i < n) p[i] = 0;
}

__global__ void k_hist(const int* __restrict__ etype, int E, int* __restrict__ counts) {
  int i = blockIdx.x * blockDim.x + threadIdx.x;
  if (i < E) atomicAdd(&counts[etype[i]], 1);
}

__global__ void k_prefix(const int* __restrict__ counts, int R,
                         int* __restrict__ offsets, int* __restrict__ tileoff) {
  if (threadIdx.x == 0 && blockIdx.x == 0) {
    int o = 0, t = 0;
    for (int r = 0; r < R; ++r) {
      offsets[r] = o;
      tileoff[r] = t;
      o += counts[r];
      t += (counts[r] + 31) >> 5;  // 32-edge super-tiles
    }
    offsets[R] = o;
    tileoff[R] = t;
  }
}

__global__ void k_scatter(const int* __restrict__ etype, int E,
                          const int* __restrict__ offsets, int* __restrict__ cursors,
                          int* __restrict__ perm) {
  int i = blockIdx.x * blockDim.x + threadIdx.x;
  if (i < E) {
    int r = etype[i];
    int p = atomicAdd(&cursors[r], 1);
    perm[offsets[r] + p] = i;
  }
}

__global__ void k_degree(const int* __restrict__ dst, int E, float* __restrict__ deg) {
  int i = blockIdx.x * blockDim.x + threadIdx.x;
  if (i < E) atomicAdd(&deg[dst[i]], 1.0f);
}

// ---------------- WMMA message kernel ----------------
// One wave32 per 32-edge super-tile (single relation). Two independent 16x128
// A tiles share every B load and run two accumulator chains:
//   D0/D1[16x128] = X0/X1[16x128] @ W_r^T, scatter-added into agg via f32 atomics.

__launch_bounds__(128)
__global__ void k_msg_wmma(const float* __restrict__ X,
                           const int* __restrict__ src,
                           const int* __restrict__ dst,
                           const float* __restrict__ Wrel,
                           const int* __restrict__ perm,
                           const int* __restrict__ counts,
                           const int* __restrict__ offsets,
                           const int* __restrict__ tileoff,
                           float* __restrict__ agg,
                           int R) {
  __shared__ v2f XsA[4][2][ATILE];  // ~66 KB: A-operand-layout staging, 2 tiles/wave
  const int wave = threadIdx.x >> 5;
  const int lane = threadIdx.x & 31;
  const int tile = blockIdx.x * 4 + wave;
  const int totalTiles = tileoff[R];
  const bool active = tile < totalTiles;  // wave-uniform

  int r = 0, lt = 0, base = 0, cnt = 0;
  if (active) {
    while (r + 1 < R && tile >= tileoff[r + 1]) ++r;
    lt = tile - tileoff[r];
    base = offsets[r];
    cnt = counts[r];
  }

  // Gather 32 node rows (float4-coalesced) and store in A-operand layout.
  // Row `row` of sub-tile t: lane supplies K-pairs kk=lane for half0 (v.xy)
  // and half1 (v.zw). Stores are bank-conflict-free.
#pragma unroll
  for (int row = 0; row < 32; ++row) {
    const int t = row >> 4;
    const int rr = row & 15;
    v4f v = {};
    int idx = lt * 32 + row;
    if (active && idx < cnt) {
      int e = perm[base + idx];
      int s = src[e];
      v = *(const v4f*)(X + (size_t)s * HD + lane * 4);
    }
    XsA[wave][t][rr * ASTRIDE + lane] = (v2f){v.x, v.y};
    XsA[wave][t][(16 + rr) * ASTRIDE + lane] = (v2f){v.z, v.w};
  }
  __syncthreads();
  if (!active) return;  // wave-uniform: EXEC all-ones for WMMA

  const int half = lane >> 4;
  const int m = lane & 15;
  const float* Wr = Wrel + (size_t)r * HD * HD;

  // Destination nodes for the C/D rows this lane owns (rows half*8+j).
  int dst0[8], dst1[8];
#pragma unroll
  for (int j = 0; j < 8; ++j) {
    int i0 = lt * 32 + half * 8 + j;
    int i1 = lt * 32 + 16 + half * 8 + j;
    dst0[j] = (i0 < cnt) ? dst[perm[base + i0]] : -1;
    dst1[j] = (i1 < cnt) ? dst[perm[base + i1]] : -1;
  }

#pragma unroll 1
  for (int nt = 0; nt < 8; ++nt) {
    const int n0 = nt * 16;
    v8f acc0 = {}, acc1 = {};
    // B[k][n] = Wr[n0+n][k]: lane m reads row (n0+m), contiguous float2 per k-step.
    const float* bp = Wr + (size_t)(n0 + m) * HD + 2 * half;
#pragma unroll
    for (int kk = 0; kk < 32; ++kk) {
      v2f b = {bp[kk * 4], bp[kk * 4 + 1]};
      v2f a0 = XsA[wave][0][lane * ASTRIDE + kk];
      v2f a1 = XsA[wave][1][lane * ASTRIDE + kk];
      acc0 = __builtin_amdgcn_wmma_f32_16x16x4_f32(false, a0, false, b,
                                                   (short)0, acc0, false, false);
      acc1 = __builtin_amdgcn_wmma_f32_16x16x4_f32(false, a1, false, b,
                                                   (short)0, acc1, false, false);
    }
#pragma unroll
    for (int j = 0; j < 8; ++j) {
      if (dst0[j] >= 0)
        atomicAdd(agg + (size_t)dst0[j] * HD + n0 + m, acc0[j]);
      if (dst1[j] >= 0)
        atomicAdd(agg + (size_t)dst1[j] * HD + n0 + m, acc1[j]);
    }
  }
}

// ---------------- WMMA self-transform + epilogue ----------------
// out = relu(X @ W_self^T + b_self + agg / max(deg,1)), 32 nodes per wave.

__launch_bounds__(128)
__global__ void k_out_wmma(const float* __restrict__ X,
                           const float* __restrict__ Wself,
                           const float* __restrict__ bself,
                           const float* __restrict__ agg,
                           const float* __restrict__ deg,
                           float* __restrict__ out,
                           int N) {
  __shared__ v2f XsA[4][2][ATILE];
  const int wave = threadIdx.x >> 5;
  const int lane = threadIdx.x & 31;
  const int tile = blockIdx.x * 4 + wave;
  const int tiles = (N + 31) >> 5;
  const bool active = tile < tiles;  // wave-uniform
  const int node0 = tile * 32;

#pragma unroll
  for (int row = 0; row < 32; ++row) {
    const int t = row >> 4;
    const int rr = row & 15;
    v4f v = {};
    int node = node0 + row;
    if (active && node < N)
      v = *(const v4f*)(X + (size_t)node * HD + lane * 4);
    XsA[wave][t][rr * ASTRIDE + lane] = (v2f){v.x, v.y};
    XsA[wave][t][(16 + rr) * ASTRIDE + lane] = (v2f){v.z, v.w};
  }
  __syncthreads();
  if (!active) return;

  const int half = lane >> 4;
  const int m = lane & 15;

#pragma unroll 1
  for (int nt = 0; nt < 8; ++nt) {
    const int n0 = nt * 16;
    v8f acc0 = {}, acc1 = {};
    const float* bp = Wself + (size_t)(n0 + m) * HD + 2 * half;
#pragma unroll
    for (int kk = 0; kk < 32; ++kk) {
      v2f b = {bp[kk * 4], bp[kk * 4 + 1]};
      v2f a0 = XsA[wave][0][lane * ASTRIDE + kk];
      v2f a1 = XsA[wave][1][lane * ASTRIDE + kk];
      acc0 = __builtin_amdgcn_wmma_f32_16x16x4_f32(false, a0, false, b,
                                                   (short)0, acc0, false, false);
      acc1 = __builtin_amdgcn_wmma_f32_16x16x4_f32(false, a1, false, b,
                                                   (short)0, acc1, false, false);
    }
    const float bias = bself[n0 + m];
#pragma unroll
    for (int j = 0; j < 8; ++j) {
      int n0j = node0 + half * 8 + j;
      int n1j = node0 + 16 + half * 8 + j;
      if (n0j < N) {
        float d = deg[n0j];
        d = d > 1.0f ? d : 1.0f;
        float v = acc0[j] + bias + agg[(size_t)n0j * HD + n0 + m] / d;
        out[(size_t)n0j * HD + n0 + m] = v > 0.0f ? v : 0.0f;
      }
      if (n1j < N) {
        float d = deg[n1j];
        d = d > 1.0f ? d : 1.0f;
        float v = acc1[j] + bias + agg[(size_t)n1j * HD + n0 + m] / d;
        out[(size_t)n1j * HD + n0 + m] = v > 0.0f ? v : 0.0f;
      }
    }
  }
}

// ---------------- launcher ----------------

extern "C" void kernel_launch(void* const* d_in, const int* in_sizes, int n_in,
                              void* d_out, int out_size, void* d_ws, size_t ws_size,
                              hipStream_t stream) {
  const float* X     = (const float*)d_in[0];
  const int*   eidx  = (const int*)d_in[1];
  const int*   etype = (const int*)d_in[2];
  const float* Wself = (const float*)d_in[3];
  const float* bself = (const float*)d_in[4];
  const float* Wrel  = (const float*)d_in[5];

  const int N = in_sizes[0] / HD;
  const int E = in_sizes[2];
  const int R = in_sizes[5] / (HD * HD);
  const int* src = eidx;       // edge_index[0, :]
  const int* dst = eidx + E;   // edge_index[1, :]

  // workspace layout
  char* w = (char*)d_ws;
  float* agg = (float*)w;  w += (size_t)N * HD * sizeof(float);
  float* deg = (float*)w;  w += (size_t)N * sizeof(float);
  int* perm  = (int*)w;    w += (size_t)E * sizeof(int);
  int* small = (int*)w;    // 128 ints of counters/offsets
  int* counts  = small;        // [R]
  int* offsets = small + 32;   // [R+1]
  int* cursors = small + 64;   // [R]
  int* tileoff = small + 96;   // [R+1]

  const int nAgg = N * HD;
  k_zero_f32<<<(nAgg + 255) / 256, 256, 0, stream>>>(agg, nAgg);
  k_zero_f32<<<(N + 255) / 256, 256, 0, stream>>>(deg, N);
  k_zero_i32<<<1, 128, 0, stream>>>(small, 128);

  k_hist<<<(E + 255) / 256, 256, 0, stream>>>(etype, E, counts);
  k_prefix<<<1, 1, 0, stream>>>(counts, R, offsets, tileoff);
  k_scatter<<<(E + 255) / 256, 256, 0, stream>>>(etype, E, offsets, cursors, perm);
  k_degree<<<(E + 255) / 256, 256, 0, stream>>>(dst, E, deg);

  const int maxTiles = (E + 31) / 32 + R;  // upper bound incl. per-relation padding
  k_msg_wmma<<<(maxTiles + 3) / 4, 128, 0, stream>>>(
      X, src, dst, Wrel, perm, counts, offsets, tileoff, agg, R);

  const int outTiles = (N + 31) / 32;
  k_out_wmma<<<(outTiles + 3) / 4, 128, 0, stream>>>(
      X, Wself, bself, agg, deg, (float*)d_out, N);
}